// SpatialAttention_8753143349390
// MI455X (gfx1250) — compile-verified
//
#include <hip/hip_runtime.h>

typedef __attribute__((ext_vector_type(2))) float v2f;
typedef __attribute__((ext_vector_type(8))) float v8f;

#define NB   64      // batch
#define HID  64      // hidden channels
#define NPIX 7396    // 86*86
#define NV4  1849    // NPIX/4 (exact)

// ---------------------------------------------------------------------------
// Kernel 1: per-batch S1 = sum(x), S2 = sum(x^2).  One block per batch.
// ws[0..63] = S1, ws[64..127] = S2
// ---------------------------------------------------------------------------
__global__ __launch_bounds__(256) void sa_batch_sums(const float* __restrict__ x,
                                                     float* __restrict__ ws) {
    const int b = blockIdx.x;
    const float4* xb = (const float4*)(x + (size_t)b * NPIX);  // 16B aligned (7396*4 % 16 == 0)
    float s1 = 0.f, s2 = 0.f;
    for (int i = threadIdx.x; i < NV4; i += 256) {
        float4 v = xb[i];
        s1 += v.x + v.y + v.z + v.w;
        s2 += v.x * v.x + v.y * v.y + v.z * v.z + v.w * v.w;
    }
    for (int mk = 1; mk < 32; mk <<= 1) {  // wave32 reduce
        s1 += __shfl_xor(s1, mk);
        s2 += __shfl_xor(s2, mk);
    }
    __shared__ float l1[8], l2[8];
    const int wave = threadIdx.x >> 5, lane = threadIdx.x & 31;
    if (lane == 0) { l1[wave] = s1; l2[wave] = s2; }
    __syncthreads();
    if (threadIdx.x == 0) {
        float t1 = 0.f, t2 = 0.f;
        for (int w = 0; w < 8; ++w) { t1 += l1[w]; t2 += l2[w]; }
        ws[b] = t1;
        ws[NB + b] = t2;
    }
}

// ---------------------------------------------------------------------------
// Kernel 2: per-batch alpha/beta via exact rank-2 scores built with
// V_WMMA_F32_16X16X4_F32 (K padded 2->4 with zeros; bit-exact fp32).
// One block (4 waves) per batch; wave i owns score rows h = 16i..16i+15.
// ws[128..191] = alpha, ws[192..255] = beta
// ---------------------------------------------------------------------------
__global__ __launch_bounds__(128) void sa_coeffs(const float* __restrict__ Wq,
                                                 const float* __restrict__ bq,
                                                 const float* __restrict__ Wk,
                                                 const float* __restrict__ bk,
                                                 const float* __restrict__ Wv,
                                                 const float* __restrict__ bv,
                                                 const float* __restrict__ Wo,
                                                 const float* __restrict__ bo,
                                                 float* __restrict__ ws) {
    const int b = blockIdx.x;
    const float S1 = ws[b];
    const float S2 = ws[NB + b];

    const int lane = threadIdx.x & 31;
    const int wv_  = threadIdx.x >> 5;   // wave id == 16-row block i
    const int col  = lane & 15;          // column within 16-wide tile
    const int half = lane >> 4;          // 0: lanes 0-15, 1: lanes 16-31
    const float msk = (half == 0) ? 1.0f : 0.0f;  // K=2,3 rows/cols of WMMA are zero

    // A-matrix (16x4): lanes 0-15 hold M=0..15; VGPR0=K0, VGPR1=K1; lanes 16-31 (K2,K3)=0
    const int hA = wv_ * 16 + col;
    const float wqh = Wq[hA] * msk;
    const float bqh = bq[hA] * msk;
    const float scale = 0.125f;          // 1/sqrt(HID)
    v2f aop;
    aop.x = (S2 * wqh + S1 * bqh) * scale;          // u[h]
    aop.y = (S1 * wqh + (float)NPIX * bqh) * scale; // v[h]

    // scores tiles: d[j] holds scores[16*wv_ + M, 16*j + N]
    v8f d[4];
#pragma unroll
    for (int j = 0; j < 4; ++j) {
        v2f bop;                          // B-matrix (4x16): row K0 = Wk, row K1 = bk
        bop.x = Wk[j * 16 + col] * msk;
        bop.y = bk[j * 16 + col] * msk;
        v8f cz = {};
        d[j] = __builtin_amdgcn_wmma_f32_16x16x4_f32(
            /*neg_a=*/false, aop, /*neg_b=*/false, bop,
            /*c_mod=*/(short)0, cz, /*reuse_a=*/false, /*reuse_b=*/false);
    }

    // Row softmax + folds: A_h = sum_g attn*wv, C_h = sum_g attn*bv.
    // C/D layout: element r of d[j] at lane L is row M=r+8*(L>=16), col N=L%16.
    const float wv0 = Wv[col],      wv1 = Wv[16 + col];
    const float wv2 = Wv[32 + col], wv3 = Wv[48 + col];
    const float bv0 = bv[col],      bv1 = bv[16 + col];
    const float bv2 = bv[32 + col], bv3 = bv[48 + col];

    __shared__ float ldsA[HID], ldsC[HID];
#pragma unroll
    for (int r = 0; r < 8; ++r) {
        float s0 = d[0][r], t1 = d[1][r], t2 = d[2][r], t3 = d[3][r];
        float m = fmaxf(fmaxf(s0, t1), fmaxf(t2, t3));
        for (int mk = 1; mk < 16; mk <<= 1) m = fmaxf(m, __shfl_xor(m, mk));
        const float e0 = __expf(s0 - m), e1 = __expf(t1 - m);
        const float e2 = __expf(t2 - m), e3 = __expf(t3 - m);
        float den = e0 + e1 + e2 + e3;
        float an  = e0 * wv0 + e1 * wv1 + e2 * wv2 + e3 * wv3;
        float cn  = e0 * bv0 + e1 * bv1 + e2 * bv2 + e3 * bv3;
        for (int mk = 1; mk < 16; mk <<= 1) {   // reduce over the 16-lane half (columns g)
            den += __shfl_xor(den, mk);
            an  += __shfl_xor(an,  mk);
            cn  += __shfl_xor(cn,  mk);
        }
        if (col == 0) {
            const int h = wv_ * 16 + r + 8 * half;
            ldsA[h] = an / den;
            ldsC[h] = cn / den;
        }
    }
    __syncthreads();
    if (threadIdx.x == 0) {
        float alpha = 0.f, beta = 0.f;
        for (int h = 0; h < HID; ++h) {
            alpha += Wo[h] * ldsA[h];
            beta  += Wo[h] * ldsC[h];
        }
        ws[2 * NB + b] = alpha;
        ws[3 * NB + b] = beta + bo[0];
    }
}

// ---------------------------------------------------------------------------
// Kernel 3: y[b,n] = alpha[b]*x[b,n] + beta[b]  (streaming, float4)
// ---------------------------------------------------------------------------
__global__ __launch_bounds__(256) void sa_apply(const float* __restrict__ x,
                                                const float* __restrict__ ws,
                                                float* __restrict__ y) {
    const int idx = blockIdx.x * 256 + threadIdx.x;   // float4 index
    if (idx < NB * NV4) {
        const int b = idx / NV4;
        const float a = ws[2 * NB + b];
        const float c = ws[3 * NB + b];
        float4 v = ((const float4*)x)[idx];
        float4 o;
        o.x = fmaf(a, v.x, c);
        o.y = fmaf(a, v.y, c);
        o.z = fmaf(a, v.z, c);
        o.w = fmaf(a, v.w, c);
        ((float4*)y)[idx] = o;
    }
}

extern "C" void kernel_launch(void* const* d_in, const int* in_sizes, int n_in,
                              void* d_out, int out_size, void* d_ws, size_t ws_size,
                              hipStream_t stream) {
    (void)in_sizes; (void)n_in; (void)out_size; (void)ws_size;
    const float* x  = (const float*)d_in[0];
    const float* Wq = (const float*)d_in[1];
    const float* bq = (const float*)d_in[2];
    const float* Wk = (const float*)d_in[3];
    const float* bk = (const float*)d_in[4];
    const float* Wv = (const float*)d_in[5];
    const float* bv = (const float*)d_in[6];
    const float* Wo = (const float*)d_in[7];
    const float* bo = (const float*)d_in[8];
    float* ws = (float*)d_ws;   // [0:64) S1, [64:128) S2, [128:192) alpha, [192:256) beta
    float* y  = (float*)d_out;

    sa_batch_sums<<<NB, 256, 0, stream>>>(x, ws);
    sa_coeffs<<<NB, 128, 0, stream>>>(Wq, bq, Wk, bk, Wv, bv, Wo, bo, ws);
    const int n4 = NB * NV4;
    sa_apply<<<(n4 + 255) / 256, 256, 0, stream>>>(x, ws, y);
}